// prob_decoding_35914516529837
// MI455X (gfx1250) — compile-verified
//
#include <hip/hip_runtime.h>
#include <hip/hip_bf16.h>

typedef __attribute__((ext_vector_type(2))) float v2f;
typedef __attribute__((ext_vector_type(4))) float v4f;
typedef __attribute__((ext_vector_type(8))) float v8f;

#define B_     64
#define ANUM   8
#define CNODE  128
#define NNODE  2048
#define G_     128
#define FSIZE  256

// d_ws float layout:
//   [0      .. 65535 ]  W1s  [A][64][128]  (cols 128..255 of W1, pre-scaled by BN s1)
//   [65536  .. 73727 ]  W2s  [A][16][64]   (pre-scaled by BN s2)
//   [73728  .. 73855 ]  bias2[A][16]       (b2*s2 + be2 - m2*s2)
//   [73856  .. 106623]  bias1[B][A][64]    ((W1[:, :128]@glb + b1)*s1 + be1 - m1*s1)
#define WS_W2S   65536
#define WS_B2    73728
#define WS_B1    73856

__device__ __forceinline__ v8f wmma_f32(v2f a, v2f b, v8f c) {
  // D = A(16x4 f32) * B(4x16 f32) + C(16x16 f32)
  return __builtin_amdgcn_wmma_f32_16x16x4_f32(
      /*neg_a=*/false, a, /*neg_b=*/false, b,
      /*c_mod=*/(short)0, c, /*reuse_a=*/false, /*reuse_b=*/false);
}

// ---------- phase 0a: BN-fold the weights ----------
__global__ void prep_weights(const float* __restrict__ W1, const float* __restrict__ g1,
                             const float* __restrict__ v1, const float* __restrict__ W2,
                             const float* __restrict__ b2, const float* __restrict__ g2,
                             const float* __restrict__ be2, const float* __restrict__ m2,
                             const float* __restrict__ v2, float* __restrict__ ws) {
  int t = blockIdx.x * blockDim.x + threadIdx.x;      // 65536 threads
  {
    int a = t >> 13;            // / (64*128)
    int m = (t >> 7) & 63;
    int c = t & 127;
    float s1 = g1[a * 64 + m] / sqrtf(v1[a * 64 + m] + 1e-5f);
    ws[t] = W1[(a * 64 + m) * 256 + 128 + c] * s1;    // node-feature half of W1
  }
  if (t < ANUM * 16 * 64) {
    int a = t >> 10;
    int o = (t >> 6) & 15;
    int c = t & 63;
    float s2 = g2[a * 16 + o] / sqrtf(v2[a * 16 + o] + 1e-5f);
    ws[WS_W2S + t] = W2[(a * 16 + o) * 64 + c] * s2;
  }
  if (t < ANUM * 16) {
    float s2 = g2[t] / sqrtf(v2[t] + 1e-5f);
    ws[WS_B2 + t] = b2[t] * s2 + be2[t] - m2[t] * s2;
  }
}

// ---------- phase 0b: fold (W1[:, :128] @ glb) + b1 + BN into one bias ----------
__global__ void prep_bias1(const float* __restrict__ W1, const float* __restrict__ glb,
                           const float* __restrict__ b1, const float* __restrict__ g1,
                           const float* __restrict__ be1, const float* __restrict__ m1,
                           const float* __restrict__ v1, float* __restrict__ ws) {
  int b = blockIdx.x >> 3, a = blockIdx.x & 7, m = threadIdx.x;   // 512 blocks x 64 thr
  const float* w = W1 + (size_t)(a * 64 + m) * 256;
  const float* gl = glb + b * G_;
  float dot = 0.f;
#pragma unroll 8
  for (int c = 0; c < 128; ++c) dot = fmaf(w[c], gl[c], dot);
  int am = a * 64 + m;
  float s1 = g1[am] / sqrtf(v1[am] + 1e-5f);
  ws[WS_B1 + (size_t)blockIdx.x * 64 + m] = s1 * (dot + b1[am]) + be1[am] - m1[am] * s1;
}

// ---------- phase 1: fused concat + GEMM chain + L2 normalize ----------
__global__ __launch_bounds__(256)
void fused_main(const float* __restrict__ nfeature, const float* __restrict__ glb,
                const float* __restrict__ W3, const float* __restrict__ b3,
                const float* __restrict__ ws, float* __restrict__ outc,
                float* __restrict__ outr) {
  __shared__ float sW2[16 * 64];        // 4 KiB  (BN-scaled W2 for this a)
  __shared__ float sH1[8 * 64 * 20];    // 40 KiB (per-wave h1 tile, stride-20 pad)
  __shared__ float sR[NNODE];           // 8 KiB  (unnormalized r)
  __shared__ float sGlb[G_];
  __shared__ float sB1[64];
  __shared__ float sB2[16];
  __shared__ float sW3[16];
  __shared__ float sRed[8];
  __shared__ float sInv;

  const int bidx = blockIdx.x;          // 512 blocks = (b, a)
  const int a = bidx & 7;
  const int b = bidx >> 3;
  const int tid = threadIdx.x;
  const int lane = tid & 31, wv = tid >> 5;
  const int hi = lane >> 4, ln = lane & 15;

  const float* nf = nfeature + (size_t)bidx * CNODE * NNODE;
  float* cat = outc + (size_t)bidx * FSIZE * NNODE;  // [256][2048] slab
  float* catn = cat + (size_t)CNODE * NNODE;         // rows 128..255 = nfeature copy
  const float* wsW1 = ws + (size_t)a * 64 * 128;     // A-matrix, cache-resident (32 KiB)

  for (int i = tid; i < 16 * 64; i += 256) sW2[i] = ws[WS_W2S + a * 1024 + i];
  for (int i = tid; i < G_; i += 256) sGlb[i] = glb[b * G_ + i];
  if (tid < 64) sB1[tid] = ws[WS_B1 + (size_t)bidx * 64 + tid];
  if (tid < 16) sB2[tid] = ws[WS_B2 + a * 16 + tid];
  if (tid < 16) sW3[tid] = W3[a * 16 + tid];
  const float b3v = b3[a];
  __syncthreads();

  // nfeat_cat rows 0..127: glb broadcast (streamed, non-temporal 16B stores)
  for (int i = tid; i < 128 * (NNODE / 4); i += 256) {
    int row = i >> 9;                 // 512 float4 per row
    int c4 = (i & 511) << 2;
    float gv = sGlb[row];
    v4f val = {gv, gv, gv, gv};
    __builtin_nontemporal_store(val, (v4f*)(cat + (size_t)row * NNODE + c4));
  }

  float* h1w = sH1 + wv * (64 * 20);

  for (int ot = 0; ot < 16; ++ot) {
    const int n0 = (ot << 7) + (wv << 4);   // this wave's 16-column slab
    const int n = n0 + ln;

    // ---- GEMM1: [64 x 128] x [128 x 16] via V_WMMA_F32_16X16X4_F32 ----
    v8f acc0 = {}, acc1 = {}, acc2 = {}, acc3 = {};
#pragma unroll 4
    for (int kk = 0; kk < 128; kk += 4) {
      const int k = kk + (hi << 1);
      // B fragment: lane = column n, vgpr v = K (k = v + 2*hi); read-once stream -> NT
      float bx = __builtin_nontemporal_load(nf + (size_t)k * NNODE + n);
      float by = __builtin_nontemporal_load(nf + (size_t)(k + 1) * NNODE + n);
      __builtin_nontemporal_store(bx, catn + (size_t)k * NNODE + n);       // fused copy
      __builtin_nontemporal_store(by, catn + (size_t)(k + 1) * NNODE + n);
      v2f bf = {bx, by};
      // A fragments: lane = row m, k = v + 2*hi (hot cached weight reads, keep RT)
      v2f a0 = *(const v2f*)(wsW1 + (0 * 16 + ln) * 128 + k);
      v2f a1 = *(const v2f*)(wsW1 + (1 * 16 + ln) * 128 + k);
      v2f a2 = *(const v2f*)(wsW1 + (2 * 16 + ln) * 128 + k);
      v2f a3 = *(const v2f*)(wsW1 + (3 * 16 + ln) * 128 + k);
      acc0 = wmma_f32(a0, bf, acc0);
      acc1 = wmma_f32(a1, bf, acc1);
      acc2 = wmma_f32(a2, bf, acc2);
      acc3 = wmma_f32(a3, bf, acc3);
    }

    // bias + ReLU, park h1 [64 x 16] in per-wave LDS (C layout: m = e + 8*hi + 16*mt)
#pragma unroll
    for (int e = 0; e < 8; ++e) {
      int m0 = (hi << 3) + e;
      h1w[(m0 +  0) * 20 + ln] = fmaxf(acc0[e] + sB1[m0 +  0], 0.f);
      h1w[(m0 + 16) * 20 + ln] = fmaxf(acc1[e] + sB1[m0 + 16], 0.f);
      h1w[(m0 + 32) * 20 + ln] = fmaxf(acc2[e] + sB1[m0 + 32], 0.f);
      h1w[(m0 + 48) * 20 + ln] = fmaxf(acc3[e] + sB1[m0 + 48], 0.f);
    }

    // ---- GEMM2: [16 x 64] x [64 x 16] ----
    v8f c2 = {};
#pragma unroll 4
    for (int kk = 0; kk < 64; kk += 4) {
      const int k = kk + (hi << 1);
      v2f af = *(const v2f*)(sW2 + ln * 64 + k);
      v2f bf = {h1w[k * 20 + ln], h1w[(k + 1) * 20 + ln]};
      c2 = wmma_f32(af, bf, c2);
    }

    // ---- GEMM3 (dot with W3) + bias2 + ReLU, fold to r[n] ----
    float part = 0.f;
#pragma unroll
    for (int e = 0; e < 8; ++e) {
      int m = (hi << 3) + e;
      float h = fmaxf(c2[e] + sB2[m], 0.f);
      part = fmaf(sW3[m], h, part);
    }
    part += __shfl_xor(part, 16, 32);   // combine m=0..7 with m=8..15 halves
    if (hi == 0) sR[n0 + ln] = part + b3v;
  }
  __syncthreads();

  // ---- L2 norm over N=2048 for this (b,a) ----
  float s = 0.f;
  for (int i = tid; i < NNODE; i += 256) { float r = sR[i]; s = fmaf(r, r, s); }
#pragma unroll
  for (int off = 16; off; off >>= 1) s += __shfl_down(s, off, 32);
  if (lane == 0) sRed[wv] = s;
  __syncthreads();
  if (tid == 0) {
    float t = 0.f;
#pragma unroll
    for (int i = 0; i < 8; ++i) t += sRed[i];
    sInv = 1.0f / fmaxf(sqrtf(t), 1e-12f);
  }
  __syncthreads();
  float* res = outr + (size_t)bidx * NNODE;
  const float inv = sInv;
  for (int i = tid; i < NNODE; i += 256) res[i] = sR[i] * inv;
}

extern "C" void kernel_launch(void* const* d_in, const int* in_sizes, int n_in,
                              void* d_out, int out_size, void* d_ws, size_t ws_size,
                              hipStream_t stream) {
  const float* nfeature = (const float*)d_in[0];
  const float* glb      = (const float*)d_in[1];
  const float* W1  = (const float*)d_in[2];
  const float* b1  = (const float*)d_in[3];
  const float* g1  = (const float*)d_in[4];
  const float* be1 = (const float*)d_in[5];
  const float* m1  = (const float*)d_in[6];
  const float* v1  = (const float*)d_in[7];
  const float* W2  = (const float*)d_in[8];
  const float* b2  = (const float*)d_in[9];
  const float* g2  = (const float*)d_in[10];
  const float* be2 = (const float*)d_in[11];
  const float* m2  = (const float*)d_in[12];
  const float* v2  = (const float*)d_in[13];
  const float* W3  = (const float*)d_in[14];
  const float* b3  = (const float*)d_in[15];
  float* ws   = (float*)d_ws;
  float* outc = (float*)d_out;                               // nfeat_cat [64,8,256,2048,1]
  float* outr = outc + (size_t)B_ * ANUM * FSIZE * NNODE;    // result    [64,8,2048]

  prep_weights<<<256, 256, 0, stream>>>(W1, g1, v1, W2, b2, g2, be2, m2, v2, ws);
  prep_bias1<<<512, 64, 0, stream>>>(W1, glb, b1, g1, be1, m1, v1, ws);
  fused_main<<<512, 256, 0, stream>>>(nfeature, glb, W3, b3, ws, outc, outr);
}